// DynamicActivation_58866821759535
// MI455X (gfx1250) — compile-verified
//
#include <hip/hip_runtime.h>

// out = tanh(x * sigmoid(x)) + sigmoid(x), elementwise fp32.
// Bandwidth-bound: 1.07 GB traffic @ 23.3 TB/s -> ~46us floor.
// Strategy: 1 float4 per lane (global_load_b128/global_store_b128),
// non-temporal cache policy (stream >> 192MB L2), TRANS-unit exp2/rcp.

typedef float f32x4 __attribute__((ext_vector_type(4)));

__device__ __forceinline__ float act(float x) {
    const float LOG2E = 1.4426950408889634f;
    // s = sigmoid(x) = 1 / (1 + 2^(-x*log2e))        -> v_exp_f32 + v_rcp_f32
    float e  = __builtin_amdgcn_exp2f(-x * LOG2E);
    float s  = __builtin_amdgcn_rcpf(1.0f + e);
    // y = x * sigmoid(x)  (SiLU/swish)
    float y  = x * s;
    // tanh(y) = 1 - 2/(2^(2y*log2e) + 1); rcp(inf)=0 gives correct +1 saturation
    float t  = __builtin_amdgcn_exp2f(y * (2.0f * LOG2E));
    float th = 1.0f - 2.0f * __builtin_amdgcn_rcpf(t + 1.0f);
    return th + s;
}

__global__ __launch_bounds__(256) void dynact_silu_tanh_kernel(
    const float* __restrict__ x, float* __restrict__ out, long long n) {
    long long i4 = ((long long)blockIdx.x * blockDim.x + threadIdx.x) * 4;
    if (i4 + 3 < n) {
        // 16B/lane -> global_load_b128 with TH=NT (streaming, bypass-friendly)
        f32x4 v = __builtin_nontemporal_load((const f32x4*)(x + i4));
        f32x4 r;
        r.x = act(v.x);
        r.y = act(v.y);
        r.z = act(v.z);
        r.w = act(v.w);
        __builtin_nontemporal_store(r, (f32x4*)(out + i4));
    } else if (i4 < n) {
        // scalar tail (n not divisible by 4)
        for (long long i = i4; i < n; ++i) {
            float v = __builtin_nontemporal_load(x + i);
            __builtin_nontemporal_store(act(v), out + i);
        }
    }
}

extern "C" void kernel_launch(void* const* d_in, const int* in_sizes, int n_in,
                              void* d_out, int out_size, void* d_ws, size_t ws_size,
                              hipStream_t stream) {
    (void)n_in; (void)d_ws; (void)ws_size; (void)out_size;
    const float* x = (const float*)d_in[0];
    float* out = (float*)d_out;
    long long n = (long long)in_sizes[0];          // 16*2048*4096 = 134,217,728

    long long nvec = (n + 3) / 4;                  // one float4 per lane
    int block = 256;                               // 8 wave32s per block
    long long grid = (nvec + block - 1) / block;   // ~131072 blocks
    dynact_silu_tanh_kernel<<<(dim3)(unsigned)grid, block, 0, stream>>>(x, out, n);
}